// RankLoss_20478404067756
// MI455X (gfx1250) — compile-verified
//
#include <hip/hip_runtime.h>
#include <stdint.h>

#define B_ROWS 32768
#define C_ENT 1000
#define C_REL 500
#define Q_ENT 250   // float4 chunks per entity row
#define Q_REL 125   // float4 chunks per relation row
#define WAVES_PER_BLOCK 8

typedef float v2f __attribute__((ext_vector_type(2)));
typedef float v8f __attribute__((ext_vector_type(8)));

struct RowStat { float m, z, v2; int i1; };

// ---- online softmax + top2 update (branchless) -----------------------------
__device__ __forceinline__ void upd(float v, int col, float& m, float& z,
                                    float& v2, int& i1) {
  float mo = m;
  bool g = v > m;                 // strict > keeps first index on ties (top_k)
  float mn = g ? v : m;
  v2 = fmaxf(v2, g ? mo : v);     // second max = max(v2, min(mo, v))
  i1 = g ? col : i1;
  z = z * __expf(mo - mn) + __expf(v - mn);
  m = mn;
}

// ---- issue async global->LDS copies for one row (Q float4 chunks) ----------
template <int Q>
__device__ __forceinline__ void async_copy_row(const float4* __restrict__ g,
                                               float4* l, int lane) {
  constexpr int ITERS = (Q + 31) >> 5;
#pragma unroll
  for (int it = 0; it < ITERS; ++it) {
    int idx = (it << 5) + lane;
    int cidx = idx < Q ? idx : (Q - 1);       // clamp tail: same src & dst -> benign dup
    uint64_t gaddr = (uint64_t)(uintptr_t)(g + cidx);
    uint32_t laddr = (uint32_t)(uintptr_t)(l + cidx);  // low 32b of generic = LDS offset
    asm volatile("global_load_async_to_lds_b128 %0, %1, off"
                 :: "v"(laddr), "v"(gaddr) : "memory");
  }
}

// ---- reduce one row held in LDS: wave-wide (m, Z, v2, i1) ------------------
template <int Q>
__device__ __forceinline__ RowStat reduce_row(const float4* lb, int lane) {
  float m = -3.0e38f, z = 0.0f, v2 = -3.0e38f;
  int i1 = 0;
  for (int q = lane; q < Q; q += 32) {
    float4 x = lb[q];             // ds_load_b128
    int c = q << 2;
    upd(x.x, c + 0, m, z, v2, i1);
    upd(x.y, c + 1, m, z, v2, i1);
    upd(x.z, c + 2, m, z, v2, i1);
    upd(x.w, c + 3, m, z, v2, i1);
  }
  // wave32 butterfly merge
#pragma unroll
  for (int off = 16; off > 0; off >>= 1) {
    float mo  = __shfl_xor(m, off, 32);
    float zo  = __shfl_xor(z, off, 32);
    float v2o = __shfl_xor(v2, off, 32);
    int   io  = __shfl_xor(i1, off, 32);
    float mn = fmaxf(m, mo);
    z  = z * __expf(m - mn) + zo * __expf(mo - mn);
    v2 = fmaxf(fmaxf(v2, v2o), fminf(m, mo));
    bool take = (mo > m) || (mo == m && io < i1);
    i1 = take ? io : i1;
    m = mn;
  }
  RowStat s; s.m = m; s.z = z; s.v2 = v2; s.i1 = i1;
  return s;
}

// ---- kernel 1: one wave per row, async double-buffered pipeline ------------
__global__ __launch_bounds__(256) void rank_rows(
    const float* __restrict__ sub, const float* __restrict__ rel,
    const float* __restrict__ obj, const int* __restrict__ st,
    const int* __restrict__ rt, const int* __restrict__ ot,
    float* __restrict__ ws) {
  __shared__ float4 lbuf[WAVES_PER_BLOCK][2][Q_ENT];   // 64000 B
  const int lane = threadIdx.x & 31;
  const int wave = threadIdx.x >> 5;
  const int row  = blockIdx.x * WAVES_PER_BLOCK + wave;

  const float4* subr = (const float4*)(sub + (size_t)row * C_ENT);
  const float4* relr = (const float4*)(rel + (size_t)row * C_REL);
  const float4* objr = (const float4*)(obj + (size_t)row * C_ENT);
  float4* b0 = &lbuf[wave][0][0];
  float4* b1 = &lbuf[wave][1][0];

  // pipeline: sub->b0 (8 ops), rel->b1 (4 ops), then obj->b0 behind sub's compute
  async_copy_row<Q_ENT>(subr, b0, lane);               // ASYNCcnt += 8
  async_copy_row<Q_REL>(relr, b1, lane);               // ASYNCcnt += 4
  asm volatile("s_wait_asynccnt 0x4" ::: "memory");    // sub landed
  RowStat S = reduce_row<Q_ENT>(b0, lane);
  // ensure our pending DS reads of b0 retired before async engine rewrites b0
  asm volatile("s_wait_dscnt 0x0" ::: "memory");
  async_copy_row<Q_ENT>(objr, b0, lane);               // ASYNCcnt += 8
  asm volatile("s_wait_asynccnt 0x8" ::: "memory");    // rel landed (in-order)
  RowStat R = reduce_row<Q_REL>(b1, lane);
  asm volatile("s_wait_asynccnt 0x0" ::: "memory");    // obj landed
  RowStat O = reduce_row<Q_ENT>(b0, lane);

  // target raw values (uniform scalar loads)
  int ts = st[row], tr = rt[row], to = ot[row];
  float xs = sub[(size_t)row * C_ENT + ts];
  float xr = rel[(size_t)row * C_REL + tr];
  float xo = obj[(size_t)row * C_ENT + to];

  // softmax-derived quantities
  float p1s = 1.0f / S.z, p2s = __expf(S.v2 - S.m) / S.z, pts = __expf(xs - S.m) / S.z;
  float p1r = 1.0f / R.z, p2r = __expf(R.v2 - R.m) / R.z, ptr_ = __expf(xr - R.m) / R.z;
  float p1o = 1.0f / O.z, p2o = __expf(O.v2 - O.m) / O.z, pto = __expf(xo - O.m) / O.z;

  bool cond = (S.i1 == ts) & (R.i1 == tr) & (O.i1 == to);
  float gt = pts * ptr_ * pto;
  float top1 = p1s * p1r * p1o;

  // second-smallest of the 8 top-2 products
  float as_[2] = {p1s, p2s}, bs_[2] = {p1r, p2r}, cs_[2] = {p1o, p2o};
  float mn1 = 3.0e38f, mn2 = 3.0e38f;
#pragma unroll
  for (int i = 0; i < 2; ++i)
#pragma unroll
    for (int j = 0; j < 2; ++j)
#pragma unroll
      for (int k = 0; k < 2; ++k) {
        float v = as_[i] * bs_[j] * cs_[k];
        if (v < mn1) { mn2 = mn1; mn1 = v; }
        else if (v < mn2) { mn2 = v; }
      }

  float pre = cond ? mn2 : top1;
  float loss = fmaxf(1.0f - gt + pre, 0.0f);
  if (lane == 0) ws[row] = loss;
}

// ---- kernel 2: deterministic mean of 32768 losses via WMMA f32 -------------
__global__ __launch_bounds__(256) void reduce_mean(const float* __restrict__ ws,
                                                   float* __restrict__ out) {
  __shared__ float wsum[WAVES_PER_BLOCK];
  const int lane = threadIdx.x & 31;
  const int wave = threadIdx.x >> 5;
  const int chunk = B_ROWS / WAVES_PER_BLOCK;   // 4096 per wave
  const int base = wave * chunk;
  float tot;
#if __has_builtin(__builtin_amdgcn_wmma_f32_16x16x4_f32)
  // A = 64 data values (16x4 f32), B = ones(4x16): D[m][n] = sum_k A[m][k] + C[m][n]
  v8f acc = {0.f, 0.f, 0.f, 0.f, 0.f, 0.f, 0.f, 0.f};
  v2f ones; ones[0] = 1.0f; ones[1] = 1.0f;
  for (int it = 0; it < chunk / 64; ++it) {
    int o = base + it * 64 + lane * 2;
    v2f a; a[0] = ws[o]; a[1] = ws[o + 1];
    acc = __builtin_amdgcn_wmma_f32_16x16x4_f32(
        false, a, false, ones, (short)0, acc, false, false);
  }
  // column N=0 lives on lane0 (M=0..7) and lane16 (M=8..15), one M per VGPR
  float s = acc[0] + acc[1] + acc[2] + acc[3] + acc[4] + acc[5] + acc[6] + acc[7];
  tot = __shfl(s, 0, 32) + __shfl(s, 16, 32);
#else
  float s = 0.0f;
  for (int i = lane; i < chunk; i += 32) s += ws[base + i];
#pragma unroll
  for (int off = 16; off > 0; off >>= 1) s += __shfl_xor(s, off, 32);
  tot = s;
#endif
  if (lane == 0) wsum[wave] = tot;
  __syncthreads();
  if (threadIdx.x == 0) {
    float t = 0.0f;
#pragma unroll
    for (int i = 0; i < WAVES_PER_BLOCK; ++i) t += wsum[i];
    *out = t / (float)B_ROWS;
  }
}

extern "C" void kernel_launch(void* const* d_in, const int* in_sizes, int n_in,
                              void* d_out, int out_size, void* d_ws, size_t ws_size,
                              hipStream_t stream) {
  (void)in_sizes; (void)n_in; (void)out_size; (void)ws_size;
  const float* sub = (const float*)d_in[0];
  const float* rel = (const float*)d_in[1];
  const float* obj = (const float*)d_in[2];
  const int* st = (const int*)d_in[3];
  const int* rt = (const int*)d_in[4];
  const int* ot = (const int*)d_in[5];
  float* ws = (float*)d_ws;
  rank_rows<<<B_ROWS / WAVES_PER_BLOCK, 256, 0, stream>>>(sub, rel, obj, st, rt, ot, ws);
  reduce_mean<<<1, 256, 0, stream>>>(ws, (float*)d_out);
}